// GravNet_simple_1271310320344
// MI455X (gfx1250) — compile-verified
//
#include <hip/hip_runtime.h>
#include <math.h>

typedef __attribute__((ext_vector_type(2))) float v2f;
typedef __attribute__((ext_vector_type(8))) float v8f;

constexpr int B = 4, V = 4096, F_IN = 64, P = 64, S = 4, F_OUT = 64;
constexpr int K = 40;                 // top-k including self
constexpr int KN = 39;                // neighbours after dropping self
constexpr int UPD = F_IN + 2 * P;     // 192

// ---------------- Kernel 1: coords = x @ Ws + bs  (S = 4) ----------------
__global__ void coords_kernel(const float* __restrict__ x,
                              const float* __restrict__ Ws,
                              const float* __restrict__ bs,
                              float4* __restrict__ coords) {
  int idx = blockIdx.x * blockDim.x + threadIdx.x;   // over B*V
  if (idx >= B * V) return;
  const float* xr = x + (size_t)idx * F_IN;
  float c0 = bs[0], c1 = bs[1], c2 = bs[2], c3 = bs[3];
#pragma unroll 8
  for (int k = 0; k < F_IN; ++k) {
    float xv = xr[k];
    const float* w = Ws + k * S;
    c0 = fmaf(xv, w[0], c0);
    c1 = fmaf(xv, w[1], c1);
    c2 = fmaf(xv, w[2], c2);
    c3 = fmaf(xv, w[3], c3);
  }
  coords[idx] = make_float4(c0, c1, c2, c3);
}

// ---------------- Kernel 2: feats = x @ Wf + bf  (WMMA fp32) ----------------
// block = 128 threads (4 waves). Wave w computes the 16x16 tile at
// rows [blockIdx.x*16, +16), cols [w*16, +16). K=64 in 16 steps of 4.
__global__ void feats_kernel(const float* __restrict__ x,
                             const float* __restrict__ Wf,
                             const float* __restrict__ bf,
                             float* __restrict__ feats) {
  const int lane = threadIdx.x & 31;
  const int wv   = threadIdx.x >> 5;
  const int half = lane >> 4;          // 0: K pair {0,1}, 1: K pair {2,3}
  const int l15  = lane & 15;          // A row (M) == B/D column (N)
  const int rowBase = blockIdx.x * 16;
  const int b = blockIdx.y;
  const int col = wv * 16 + l15;

  const float* xrow = x + ((size_t)b * V + rowBase + l15) * F_IN + half * 2;
  v8f acc = {};
#pragma unroll
  for (int k = 0; k < F_IN; k += 4) {
    v2f a = *(const v2f*)(xrow + k);             // A[m][k+2h], A[m][k+2h+1]
    v2f bm;
    bm.x = Wf[(size_t)(k + half * 2) * P + col]; // B[k+2h][n]
    bm.y = Wf[(size_t)(k + half * 2 + 1) * P + col];
    acc = __builtin_amdgcn_wmma_f32_16x16x4_f32(false, a, false, bm,
                                                (short)0, acc, false, false);
  }
  const float bias = bf[col];
  float* orow = feats + ((size_t)b * V + rowBase + half * 8) * P + col;
#pragma unroll
  for (int r = 0; r < 8; ++r)
    orow[(size_t)r * P] = acc[r] + bias;
}

// ---------------- Kernel 3: brute-force KNN (top-40 smallest d2) ----------------
// 64 query vertices per block; candidate coords staged in LDS in 1024-point
// tiles (16 KB); per-thread sorted top-40 list in LDS (20.5 KB).
__global__ void knn_kernel(const float4* __restrict__ coords,
                           float* __restrict__ nbr_d,
                           int* __restrict__ nbr_i) {
  __shared__ float4 cT[1024];
  __shared__ float hd[64 * K];
  __shared__ int   hi[64 * K];

  const int t = threadIdx.x;                  // 64 threads
  const int b = blockIdx.y;
  const int v = blockIdx.x * 64 + t;
  const float4 cq = coords[(size_t)b * V + v];

  float* myd = hd + t * K;
  int*   myi = hi + t * K;
#pragma unroll
  for (int k = 0; k < K; ++k) { myd[k] = 3.4e38f; myi[k] = 0; }
  float worst = 3.4e38f;

  for (int tile = 0; tile < V / 1024; ++tile) {
    __syncthreads();
    for (int j = t; j < 1024; j += 64)
      cT[j] = coords[(size_t)b * V + tile * 1024 + j];
    __syncthreads();
    for (int u = 0; u < 1024; ++u) {
      float4 cu = cT[u];
      float dx = cq.x - cu.x, dy = cq.y - cu.y;
      float dz = cq.z - cu.z, dw = cq.w - cu.w;
      float d2 = dx * dx + dy * dy + dz * dz + dw * dw;
      if (d2 < worst) {                       // sorted insertion
        int pos = K - 1;
        while (pos > 0 && myd[pos - 1] > d2) {
          myd[pos] = myd[pos - 1];
          myi[pos] = myi[pos - 1];
          --pos;
        }
        myd[pos] = d2;
        myi[pos] = tile * 1024 + u;
        worst = myd[K - 1];
      }
    }
  }
  float* od = nbr_d + ((size_t)b * V + v) * K;
  int*   oi = nbr_i + ((size_t)b * V + v) * K;
#pragma unroll
  for (int k = 0; k < K; ++k) { od[k] = myd[k]; oi[k] = myi[k]; }
}

// ---------------- Kernel 4: gather + weight + max/mean, build `updated` ----------------
// 1 wave per vertex (8 waves/block); each lane owns 2 feature channels.
__global__ void gather_kernel(const float* __restrict__ x,
                              const float* __restrict__ feats,
                              const float* __restrict__ nbr_d,
                              const int* __restrict__ nbr_i,
                              float* __restrict__ upd) {
  const int lane = threadIdx.x & 31;
  const int wv   = threadIdx.x >> 5;
  const int b = blockIdx.y;
  const int v = blockIdx.x * 8 + wv;
  const size_t bv = (size_t)b * V + v;

  const float* dlist = nbr_d + bv * K;
  const int*   ilist = nbr_i + bv * K;

  float mx0 = -3.4e38f, mx1 = -3.4e38f, s0 = 0.f, s1 = 0.f;
  for (int j = 1; j < K; ++j) {               // skip self at position 0
    const int ui = ilist[j];
    const float w = __expf(-fabsf(dlist[j] * 10.0f));
    v2f f = *(const v2f*)(feats + ((size_t)b * V + ui) * P + 2 * lane);
    float v0 = f.x * w, v1 = f.y * w;
    mx0 = fmaxf(mx0, v0); mx1 = fmaxf(mx1, v1);
    s0 += v0; s1 += v1;
  }
  float* ur = upd + bv * UPD;
  v2f xr = *(const v2f*)(x + bv * F_IN + 2 * lane);
  *(v2f*)(ur + 2 * lane) = xr;                       // [0:64)   x
  v2f mx; mx.x = mx0; mx.y = mx1;
  *(v2f*)(ur + F_IN + 2 * lane) = mx;                // [64:128) max-pool
  v2f mn; mn.x = s0 * (1.0f / KN); mn.y = s1 * (1.0f / KN);
  *(v2f*)(ur + F_IN + P + 2 * lane) = mn;            // [128:192) mean-pool
}

// ---------------- Kernel 5: out = tanh(updated @ Wo + bo)  (WMMA fp32) ----------------
// Rows flattened over B*V = 16384; K=192 -> 48 wmma steps; 4 waves cover N=64.
__global__ void out_kernel(const float* __restrict__ upd,
                           const float* __restrict__ Wo,
                           const float* __restrict__ bo,
                           float* __restrict__ out) {
  const int lane = threadIdx.x & 31;
  const int wv   = threadIdx.x >> 5;
  const int half = lane >> 4;
  const int l15  = lane & 15;
  const int rowBase = blockIdx.x * 16;
  const int col = wv * 16 + l15;

  const float* arow = upd + ((size_t)rowBase + l15) * UPD + half * 2;
  v8f acc = {};
#pragma unroll
  for (int k = 0; k < UPD; k += 4) {
    v2f a = *(const v2f*)(arow + k);
    v2f bm;
    bm.x = Wo[(size_t)(k + half * 2) * F_OUT + col];
    bm.y = Wo[(size_t)(k + half * 2 + 1) * F_OUT + col];
    acc = __builtin_amdgcn_wmma_f32_16x16x4_f32(false, a, false, bm,
                                                (short)0, acc, false, false);
  }
  const float bias = bo[col];
  float* orow = out + ((size_t)rowBase + half * 8) * F_OUT + col;
#pragma unroll
  for (int r = 0; r < 8; ++r)
    orow[(size_t)r * F_OUT] = tanhf(acc[r] + bias);
}

extern "C" void kernel_launch(void* const* d_in, const int* in_sizes, int n_in,
                              void* d_out, int out_size, void* d_ws, size_t ws_size,
                              hipStream_t stream) {
  const float* x  = (const float*)d_in[0];
  const float* Wf = (const float*)d_in[1];
  const float* bf = (const float*)d_in[2];
  const float* Ws = (const float*)d_in[3];
  const float* bs = (const float*)d_in[4];
  const float* Wo = (const float*)d_in[5];
  const float* bo = (const float*)d_in[6];
  float* out = (float*)d_out;

  // Workspace layout (bytes):
  //   coords : B*V*4  f32 =   256 KB @ 0
  //   feats  : B*V*64 f32 =  4096 KB @ 262144
  //   nbr_d  : B*V*40 f32 =  2560 KB @ 4456448
  //   nbr_i  : B*V*40 i32 =  2560 KB @ 7077888
  //   upd    : B*V*192 f32 = 12288 KB @ 9699328   (total ~21.3 MB)
  char* ws = (char*)d_ws;
  float4* coords = (float4*)(ws);
  float*  feats  = (float*)(ws + 262144);
  float*  nbr_d  = (float*)(ws + 4456448);
  int*    nbr_i  = (int*)(ws + 7077888);
  float*  upd    = (float*)(ws + 9699328);

  coords_kernel<<<dim3((B * V + 255) / 256), 256, 0, stream>>>(x, Ws, bs, coords);
  feats_kernel <<<dim3(V / 16, B), 128, 0, stream>>>(x, Wf, bf, feats);
  knn_kernel   <<<dim3(V / 64, B), 64, 0, stream>>>(coords, nbr_d, nbr_i);
  gather_kernel<<<dim3(V / 8, B), 256, 0, stream>>>(x, feats, nbr_d, nbr_i, upd);
  out_kernel   <<<dim3((B * V) / 16), 128, 0, stream>>>(upd, Wo, bo, out);
}